// CRF_12060268167595
// MI455X (gfx1250) — compile-verified
//
#include <hip/hip_runtime.h>

// Linear-chain CRF NLL, fully WMMA-based:
//   emissions  : per-position 16x128 * 128x32 f32-WMMA GEMM (rows = 16 words/wave)
//   recurrence : alpha' = m + log( exp(alpha+e-m) x exp(T) )  -> 16x32 * 32x32 WMMA
// Wave = 16 words; C-layout (row=word, col=label) flows through the whole chain.

static constexpr int Mlen = 16;    // word length
static constexpr int Dft  = 128;   // feature dim
static constexpr int Lbl  = 26;    // labels
static constexpr int Lpad = 32;    // labels padded to two 16-col WMMA tiles
static constexpr int ELS  = 33;    // E-tile LDS row stride (padded, conflict-free)
static constexpr int WPB  = 4;     // waves per block
static constexpr int THREADS = WPB * 32;
static constexpr int GW   = 16;    // words per wave

typedef __attribute__((ext_vector_type(2))) float v2f;
typedef __attribute__((ext_vector_type(8))) float v8f;

__global__ __launch_bounds__(THREADS)
void crf_main(const float* __restrict__ X, const int* __restrict__ Y,
              const float* __restrict__ W, const float* __restrict__ T,
              float* __restrict__ partials, int nWords)
{
    __shared__ float Wp[Lpad * Dft];        // 16KB: W zero-padded to 32 rows, [c][k]
    __shared__ float expTt[Lpad * Lpad];    // 4KB : expTt[n][k] = exp(T[k][n]), padded 0
    __shared__ float Traw[864];             // raw T row-major + zero slack for masked idx
    __shared__ float Psh [WPB][GW * Lpad];  // per-wave P transpose buffer
    __shared__ float ELsh[WPB][GW * ELS];   // per-wave emission tile (node gather)
    __shared__ int   Ysh [WPB][GW * Mlen];  // per-wave label matrix
    __shared__ float LZsh[WPB][GW];
    __shared__ float wpart[WPB];

    const int tid = threadIdx.x;

    // ---- stage shared operands ----
    for (int idx = tid; idx < Lpad * Dft; idx += THREADS) {
        int c = idx >> 7, k = idx & (Dft - 1);
        Wp[idx] = (c < Lbl) ? W[c * Dft + k] : 0.f;
    }
    for (int idx = tid; idx < Lpad * Lpad; idx += THREADS) {
        int n = idx >> 5, k = idx & (Lpad - 1);
        expTt[idx] = (n < Lbl && k < Lbl) ? __expf(T[k * Lbl + n]) : 0.f;
    }
    for (int idx = tid; idx < 864; idx += THREADS)
        Traw[idx] = (idx < Lbl * Lbl) ? T[idx] : 0.f;
    __syncthreads();

    const int wave = tid >> 5;
    const int lane = tid & 31;
    const int lo   = lane & 15;
    const int hi   = lane >> 4;
    const int base = (blockIdx.x * WPB + wave) * GW;   // first word of this wave's group
    float res = 0.f;

    if (base < nWords) {   // wave-uniform
        // A-operand row base: lane lo -> word base+lo (clamped), K-pair offset by hi
        int wclamp = base + lo; if (wclamp > nWords - 1) wclamp = nWords - 1;
        const float* rowbase = X + (size_t)wclamp * (Mlen * Dft);   // lane's word row 0
        const float* arow0   = rowbase + (hi << 1);
        const float* b0p     = &Wp[lo * Dft + (hi << 1)];           // emission B, cols 0..15
        const float* b1p     = &Wp[(16 + lo) * Dft + (hi << 1)];    // emission B, cols 16..31

        // preload exp(T) B-fragments into registers (constant across words/steps)
        v2f bt0[8], bt1[8];
        #pragma unroll
        for (int kk = 0; kk < 8; ++kk) {
            bt0[kk] = *(const v2f*)&expTt[lo * Lpad + (kk << 2) + (hi << 1)];
            bt1[kk] = *(const v2f*)&expTt[(16 + lo) * Lpad + (kk << 2) + (hi << 1)];
        }

        // stage this wave's Y block (256 ints, coalesced, clamped)
        int* Yw = Ysh[wave];
        const int lastY = nWords * Mlen - 1;
        #pragma unroll
        for (int i = 0; i < 8; ++i) {
            int idx = lane + (i << 5);
            int g = base * Mlen + idx; if (g > lastY) g = lastY;
            Yw[idx] = Y[g];
        }

        float* Pb  = Psh[wave];
        float* ELb = ELsh[wave];
        const bool padHi = (lo >= (Lbl - 16));   // col 16+lo >= 26 -> pad label
        const float gmask = (lane < GW && (base + lane) < nWords) ? 1.f : 0.f;

        v8f a0 = {}; v8f a1 = {};                // alpha, C-layout
        float node = 0.f, edge = 0.f;
        int yprev = 0;

        for (int s = 0; s < Mlen; ++s) {
            // prefetch next position's rows (4 x 128B lines, split across halves)
            if (s + 1 < Mlen) {
                const float* prow = rowbase + (s + 1) * Dft + (hi << 5);
                __builtin_prefetch(prow, 0, 3);
                __builtin_prefetch(prow + 64, 0, 3);
            }

            // ===== emissions for position s: E[16 words x 32 labels] =====
            // 4 independent accumulation chains (2 N-tiles x 2 K-halves) for ILP
            v8f e0a = {}; v8f e1a = {}; v8f e0b = {}; v8f e1b = {};
            const float* ar = arow0 + s * Dft;
            #pragma unroll 8
            for (int k0 = 0; k0 < Dft / 2; k0 += 4) {
                v2f a  = *(const v2f*)(ar + k0);
                v2f b0 = *(const v2f*)(b0p + k0);
                v2f b1 = *(const v2f*)(b1p + k0);
                e0a = __builtin_amdgcn_wmma_f32_16x16x4_f32(false, a, false, b0,
                                                            (short)0, e0a, false, false);
                e1a = __builtin_amdgcn_wmma_f32_16x16x4_f32(false, a, false, b1,
                                                            (short)0, e1a, false, false);
            }
            #pragma unroll 8
            for (int k0 = Dft / 2; k0 < Dft; k0 += 4) {
                v2f a  = *(const v2f*)(ar + k0);
                v2f b0 = *(const v2f*)(b0p + k0);
                v2f b1 = *(const v2f*)(b1p + k0);
                e0b = __builtin_amdgcn_wmma_f32_16x16x4_f32(false, a, false, b0,
                                                            (short)0, e0b, false, false);
                e1b = __builtin_amdgcn_wmma_f32_16x16x4_f32(false, a, false, b1,
                                                            (short)0, e1b, false, false);
            }
            v8f e0 = e0a + e0b;
            v8f e1 = e1a + e1b;

            // spill E tile for the node-potential gather
            #pragma unroll
            for (int r = 0; r < 8; ++r) {
                int row = r + (hi << 3);
                ELb[row * ELS + lo]      = e0[r];
                ELb[row * ELS + 16 + lo] = e1[r];
            }

            v8f v0 = a0 + e0;
            v8f v1 = a1 + e1;

            if (s < Mlen - 1) {
                // ===== scaled-forward step =====
                float m[8];
                #pragma unroll
                for (int r = 0; r < 8; ++r) {        // row max over valid labels
                    float loc = fmaxf(v0[r], padHi ? -1e30f : v1[r]);
                    loc = fmaxf(loc, __shfl_xor(loc, 8, 32));
                    loc = fmaxf(loc, __shfl_xor(loc, 4, 32));
                    loc = fmaxf(loc, __shfl_xor(loc, 2, 32));
                    loc = fmaxf(loc, __shfl_xor(loc, 1, 32));
                    m[r] = loc;
                }
                #pragma unroll
                for (int r = 0; r < 8; ++r) {        // P = exp(v-m), pad labels -> 0
                    int row = r + (hi << 3);
                    float p0 = __expf(v0[r] - m[r]);
                    float p1 = padHi ? 0.f : __expf(v1[r] - m[r]);
                    Pb[row * Lpad + lo]      = p0;
                    Pb[row * Lpad + 16 + lo] = p1;
                }
                // S = P(16x32) * expT(32x32) on the matrix unit
                v8f s0 = {}; v8f s1 = {};
                const float* Prow = &Pb[lo * Lpad + (hi << 1)];
                #pragma unroll
                for (int kk = 0; kk < 8; ++kk) {
                    v2f a = *(const v2f*)(Prow + (kk << 2));
                    s0 = __builtin_amdgcn_wmma_f32_16x16x4_f32(false, a, false, bt0[kk],
                                                               (short)0, s0, false, false);
                    s1 = __builtin_amdgcn_wmma_f32_16x16x4_f32(false, a, false, bt1[kk],
                                                               (short)0, s1, false, false);
                }
                #pragma unroll
                for (int r = 0; r < 8; ++r) {        // alpha = m + log(S)
                    a0[r] = m[r] + __logf(s0[r]);
                    a1[r] = padHi ? -1e30f : (m[r] + __logf(s1[r]));
                }
            } else {
                // ===== final step: log_z per word = row-lse of alpha + e =====
                float m[8];
                #pragma unroll
                for (int r = 0; r < 8; ++r) {
                    float loc = fmaxf(v0[r], padHi ? -1e30f : v1[r]);
                    loc = fmaxf(loc, __shfl_xor(loc, 8, 32));
                    loc = fmaxf(loc, __shfl_xor(loc, 4, 32));
                    loc = fmaxf(loc, __shfl_xor(loc, 2, 32));
                    loc = fmaxf(loc, __shfl_xor(loc, 1, 32));
                    m[r] = loc;
                }
                float lzr[8];
                #pragma unroll
                for (int r = 0; r < 8; ++r) {
                    float se = __expf(v0[r] - m[r]) + (padHi ? 0.f : __expf(v1[r] - m[r]));
                    se += __shfl_xor(se, 8, 32);
                    se += __shfl_xor(se, 4, 32);
                    se += __shfl_xor(se, 2, 32);
                    se += __shfl_xor(se, 1, 32);
                    lzr[r] = m[r] + __logf(se);
                }
                if (lo == 0) {
                    #pragma unroll
                    for (int r = 0; r < 8; ++r) LZsh[wave][(hi << 3) + r] = lzr[r];
                }
            }

            // ===== node + edge gathers (lane lo = word; hi-half duplicates, masked) =====
            int ys = Yw[lo * Mlen + s] & 31;
            node += ELb[lo * ELS + ys];
            if (s > 0) edge += Traw[yprev * Lbl + ys];
            yprev = ys;
        }

        float contrib = gmask * (LZsh[wave][lo] - node - edge);  // per-word -LL term
        contrib += __shfl_xor(contrib, 16, 32);
        contrib += __shfl_xor(contrib, 8, 32);
        contrib += __shfl_xor(contrib, 4, 32);
        contrib += __shfl_xor(contrib, 2, 32);
        contrib += __shfl_xor(contrib, 1, 32);
        res = contrib;
    }

    if (lane == 0) wpart[wave] = res;
    __syncthreads();
    if (tid == 0) {
        float s = 0.f;
        #pragma unroll
        for (int w = 0; w < WPB; ++w) s += wpart[w];
        partials[blockIdx.x] = s;
    }
}

// Deterministic fixed-order final reduction (no float atomics).
__global__ __launch_bounds__(256)
void crf_reduce(const float* __restrict__ partials, int n, float* __restrict__ out)
{
    __shared__ float sh[256];
    float s = 0.f;
    for (int i = threadIdx.x; i < n; i += 256) s += partials[i];
    sh[threadIdx.x] = s;
    __syncthreads();
    for (int off = 128; off > 0; off >>= 1) {
        if ((int)threadIdx.x < off) sh[threadIdx.x] += sh[threadIdx.x + off];
        __syncthreads();
    }
    if (threadIdx.x == 0) out[0] = sh[0];
}

extern "C" void kernel_launch(void* const* d_in, const int* in_sizes, int n_in,
                              void* d_out, int out_size, void* d_ws, size_t ws_size,
                              hipStream_t stream)
{
    const float* X = (const float*)d_in[0];   // [B, 16, 128] f32
    const int*   Y = (const int*)d_in[1];     // [B, 16] int
    const float* W = (const float*)d_in[2];   // [26, 128] f32
    const float* T = (const float*)d_in[3];   // [26, 26] f32

    const int nWords = in_sizes[0] / (Mlen * Dft);
    const int blocks = (nWords + WPB * GW - 1) / (WPB * GW);

    float* partials = (float*)d_ws;

    crf_main<<<blocks, THREADS, 0, stream>>>(X, Y, W, T, partials, nWords);
    crf_reduce<<<1, 256, 0, stream>>>(partials, blocks, (float*)d_out);
}